// attention_TP_Title_66795331387725
// MI455X (gfx1250) — compile-verified
//
#include <hip/hip_runtime.h>
#include <hip/hip_bf16.h>

typedef __attribute__((ext_vector_type(2))) float v2f;
typedef __attribute__((ext_vector_type(4))) float v4f;
typedef __attribute__((ext_vector_type(8))) float v8f;

#define B_    64
#define L_    4096
#define D_    128
#define NSEG  16
#define LSEG  (L_ / NSEG)   /* 256 */
#define NEG_BIG -1e9f

// Branch-free tanh: tanh(x) = 1 - 2/(exp(2x)+1).
// exp(2x)->inf => 1 ; exp(2x)->0 => -1. Single v_exp_f32 + v_rcp_f32,
// keeps EXEC uniform inside the WMMA loop (no libm divergent paths).
__device__ __forceinline__ float fast_tanh(float t) {
  float e = __expf(2.0f * t);
  return 1.0f - 2.0f * __builtin_amdgcn_rcpf(e + 1.0f);
}

// ---------------------------------------------------------------------------
// K1: scores[b,l] = sum_d y[b,l,d] * tanh(x[b,d]*w[l,d] + bias[d])
// One block per (b, L-segment). Each wave computes 16x16 score tiles with
// V_WMMA_F32_16X16X4_F32 (D = Y_tile x Qi_tile^T), keeps the diagonal.
// ---------------------------------------------------------------------------
__global__ __launch_bounds__(256) void k1_scores(
    const float* __restrict__ x, const float* __restrict__ y,
    const float* __restrict__ w, const float* __restrict__ bias,
    float* __restrict__ scores)
{
  const int b   = blockIdx.x / NSEG;
  const int seg = blockIdx.x % NSEG;
  const int l0s = seg * LSEG;

  __shared__ float xs[D_];
  __shared__ float bs[D_];
  if (threadIdx.x < D_) {
    xs[threadIdx.x] = x[b * D_ + threadIdx.x];
    bs[threadIdx.x] = bias[threadIdx.x];
  }
  __syncthreads();

  const int wave  = threadIdx.x >> 5;      // 8 waves
  const int lane  = threadIdx.x & 31;
  const int rsel  = lane & 15;             // M for A operand, N for B operand
  const int dhalf = (lane >> 4) << 1;      // lanes 0-15: K=0,1 ; lanes 16-31: K=2,3

  // 16 tiles of 16 l's per segment; 8 waves -> 2 tiles per wave (uniform EXEC)
  for (int t = wave; t < LSEG / 16; t += 8) {
    const int l0  = l0s + t * 16;
    const int row = l0 + rsel;
    const float* __restrict__ yrow = y + ((size_t)b * L_ + row) * (size_t)D_;
    const float* __restrict__ wrow = w + (size_t)row * (size_t)D_;

    v8f c = {0.f, 0.f, 0.f, 0.f, 0.f, 0.f, 0.f, 0.f};

    #pragma unroll
    for (int d0 = 0; d0 < D_; d0 += 4) {
      const int d = d0 + dhalf;            // even => 8-byte aligned
      v2f a  = *(const v2f*)(yrow + d);    // global_load_b64
      v2f wv = *(const v2f*)(wrow + d);    // global_load_b64
      v2f xv = *(const v2f*)(&xs[d]);      // ds_load_b64
      v2f bv = *(const v2f*)(&bs[d]);      // ds_load_b64
      v2f q;
      q.x = fast_tanh(__builtin_fmaf(xv.x, wv.x, bv.x));
      q.y = fast_tanh(__builtin_fmaf(xv.y, wv.y, bv.y));
      // A[m,k] = y[l0+m, d0+k], B[k,n] = qi[l0+n, d0+k]  ->  D[m,n] += Y.Qi^T
      c = __builtin_amdgcn_wmma_f32_16x16x4_f32(
              false, a, false, q, (short)0, c, false, false);
    }

    // C VGPR v holds (M=v, N=lane) lanes 0-15 and (M=8+v, N=lane-16) lanes 16-31
    // => diag (v,v) @ lane v ; (8+v,8+v) @ lane 24+v
    float diag = 0.f;
    #pragma unroll
    for (int v = 0; v < 8; ++v)
      if (lane == v || lane == 24 + v) diag = c[v];

    if (lane < 8)
      scores[(size_t)b * L_ + l0 + lane] = diag;
    else if (lane >= 24)
      scores[(size_t)b * L_ + l0 + (lane - 16)] = diag;
  }
}

// ---------------------------------------------------------------------------
// K2: masked softmax over L, in place (scores -> probabilities). 1 block / b.
// ---------------------------------------------------------------------------
__device__ __forceinline__ float wave_max(float v) {
  #pragma unroll
  for (int o = 16; o > 0; o >>= 1) v = fmaxf(v, __shfl_xor(v, o, 32));
  return v;
}
__device__ __forceinline__ float wave_sum(float v) {
  #pragma unroll
  for (int o = 16; o > 0; o >>= 1) v += __shfl_xor(v, o, 32);
  return v;
}

__global__ __launch_bounds__(256) void k2_softmax(float* __restrict__ scores)
{
  const int b = blockIdx.x;
  float* __restrict__ s = scores + (size_t)b * L_;
  __shared__ float red[8];
  const int wv = threadIdx.x >> 5;
  const int ln = threadIdx.x & 31;

  // pass 1: max of (ai + mask)
  float m = -INFINITY;
  for (int l = threadIdx.x; l < L_; l += 256) {
    float v = s[l];
    v = (v == 0.0f) ? (v + NEG_BIG) : v;   // PAD mask (branch-free cndmask)
    m = fmaxf(m, v);
  }
  m = wave_max(m);
  if (ln == 0) red[wv] = m;
  __syncthreads();
  float gm = red[0];
  #pragma unroll
  for (int i = 1; i < 8; ++i) gm = fmaxf(gm, red[i]);
  __syncthreads();

  // pass 2: sum of exp
  float sum = 0.f;
  for (int l = threadIdx.x; l < L_; l += 256) {
    float v = s[l];
    v = (v == 0.0f) ? (v + NEG_BIG) : v;
    sum += __expf(v - gm);
  }
  sum = wave_sum(sum);
  if (ln == 0) red[wv] = sum;
  __syncthreads();
  float gs = 0.f;
  #pragma unroll
  for (int i = 0; i < 8; ++i) gs += red[i];
  const float inv = 1.0f / gs;

  // pass 3: write probabilities
  for (int l = threadIdx.x; l < L_; l += 256) {
    float v = s[l];
    v = (v == 0.0f) ? (v + NEG_BIG) : v;
    s[l] = __expf(v - gm) * inv;
  }
}

// ---------------------------------------------------------------------------
// K3: partial weighted sums. One block per (b, segment).
// Each lane streams float4 columns (global_load_b128, one 512B row per
// wave-iteration) + global_prefetch_b8 8 rows ahead.
// partial[(b*NSEG+seg)*D + d] = sum_{l in seg} p[b,l] * y[b,l,d]
// ---------------------------------------------------------------------------
__global__ __launch_bounds__(256) void k3_partial(
    const float* __restrict__ y, const float* __restrict__ p,
    float* __restrict__ partial)
{
  const int b   = blockIdx.x / NSEG;
  const int seg = blockIdx.x % NSEG;
  const int l0  = seg * LSEG;

  __shared__ float ps[LSEG];
  for (int i = threadIdx.x; i < LSEG; i += 256)
    ps[i] = p[(size_t)b * L_ + l0 + i];
  __syncthreads();

  const int q = (threadIdx.x & 31) << 2;     // d-quad base: 0,4,...,124
  const int h = threadIdx.x >> 5;            // 8 L-slices of 32 rows each
  const float* __restrict__ yb = y + ((size_t)b * L_ + l0) * (size_t)D_;

  v4f acc = {0.f, 0.f, 0.f, 0.f};
  const int lbeg = h * (LSEG / 8), lend = lbeg + (LSEG / 8);
  for (int l = lbeg; l < lend; ++l) {
    __builtin_prefetch(yb + (size_t)(l + 8) * D_ + q, 0, 0);  // global_prefetch_b8
    v4f yv = *(const v4f*)(yb + (size_t)l * D_ + q);          // global_load_b128
    acc = acc + yv * ps[l];
  }

  __shared__ float comb[8][D_];              // 4 KB
  *(v4f*)&comb[h][q] = acc;
  __syncthreads();
  if (threadIdx.x < D_) {
    float sacc = 0.f;
    #pragma unroll
    for (int i = 0; i < 8; ++i) sacc += comb[i][threadIdx.x];
    partial[((size_t)b * NSEG + seg) * D_ + threadIdx.x] = sacc;
  }
}

// ---------------------------------------------------------------------------
// K4: reduce NSEG partials -> out[b,d]
// ---------------------------------------------------------------------------
__global__ __launch_bounds__(128) void k4_reduce(
    const float* __restrict__ partial, float* __restrict__ out)
{
  const int b = blockIdx.x;
  const int d = threadIdx.x;
  float acc = 0.f;
  #pragma unroll
  for (int s = 0; s < NSEG; ++s)
    acc += partial[((size_t)b * NSEG + s) * D_ + d];
  out[(size_t)b * D_ + d] = acc;
}

// ---------------------------------------------------------------------------
extern "C" void kernel_launch(void* const* d_in, const int* in_sizes, int n_in,
                              void* d_out, int out_size, void* d_ws, size_t ws_size,
                              hipStream_t stream)
{
  const float* x    = (const float*)d_in[0];   // [B,1,D]
  const float* y    = (const float*)d_in[1];   // [B,L,D]
  const float* w    = (const float*)d_in[2];   // [1,L,D]
  const float* bias = (const float*)d_in[3];   // [1,1,D]
  float* out = (float*)d_out;                  // [B,D]

  float* scores  = (float*)d_ws;                      // B*L floats   (1 MB)
  float* partial = scores + (size_t)B_ * L_;          // B*NSEG*D     (512 KB)

  hipLaunchKernelGGL(k1_scores,  dim3(B_ * NSEG), dim3(256), 0, stream,
                     x, y, w, bias, scores);
  hipLaunchKernelGGL(k2_softmax, dim3(B_),        dim3(256), 0, stream, scores);
  hipLaunchKernelGGL(k3_partial, dim3(B_ * NSEG), dim3(256), 0, stream,
                     y, scores, partial);
  hipLaunchKernelGGL(k4_reduce,  dim3(B_),        dim3(128), 0, stream,
                     partial, out);
}